// CausalLayerNorm_2061584302896
// MI455X (gfx1250) — compile-verified
//
#include <hip/hip_runtime.h>
#include <math.h>

typedef __attribute__((ext_vector_type(2))) float v2f;
typedef __attribute__((ext_vector_type(8))) float v8f;

#define B_      8
#define T_      4096
#define C_      512
#define NCHUNK  16
#define CHUNK_T (T_ / NCHUNK)   // 256
#define EPSV    1e-5f

// ---------------------------------------------------------------------------
// Pass 1: per-(b,chunk,channel) totals of x and x^2. Fully coalesced stream.
// grid = B_*NCHUNK blocks, block = C_ threads (one thread per channel).
// ---------------------------------------------------------------------------
__global__ void __launch_bounds__(C_)
cln_partials(const float* __restrict__ x,
             float* __restrict__ pS, float* __restrict__ pQ) {
    int blk = blockIdx.x;              // b*NCHUNK + k
    int c   = threadIdx.x;             // channel
    int b   = blk / NCHUNK;
    int k   = blk % NCHUNK;
    const float* xp = x + ((size_t)(b * T_ + k * CHUNK_T)) * C_ + c;
    float s = 0.0f, q = 0.0f;
    #pragma unroll 8
    for (int t = 0; t < CHUNK_T; ++t) {
        float v = xp[(size_t)t * C_];
        s += v;
        q = fmaf(v, v, q);
    }
    pS[blk * C_ + c] = s;
    pQ[blk * C_ + c] = q;
}

// ---------------------------------------------------------------------------
// Pass 2: in-place exclusive scan over the NCHUNK chunk totals per (b,c).
// B_*C_ = 4096 threads total; coalesced over c.
// ---------------------------------------------------------------------------
__global__ void __launch_bounds__(256)
cln_prefix(float* __restrict__ pS, float* __restrict__ pQ) {
    int tid = blockIdx.x * blockDim.x + threadIdx.x;   // 0 .. B_*C_-1
    int b = tid / C_;
    int c = tid % C_;
    float sv[NCHUNK], qv[NCHUNK];
    #pragma unroll
    for (int k = 0; k < NCHUNK; ++k) {
        sv[k] = pS[(b * NCHUNK + k) * C_ + c];
        qv[k] = pQ[(b * NCHUNK + k) * C_ + c];
    }
    float s = 0.0f, q = 0.0f;
    #pragma unroll
    for (int k = 0; k < NCHUNK; ++k) {
        pS[(b * NCHUNK + k) * C_ + c] = s;
        pQ[(b * NCHUNK + k) * C_ + c] = q;
        s += sv[k];
        q += qv[k];
    }
}

// ---------------------------------------------------------------------------
// Pass 3: WMMA chunked scan + normalize.
// Each wave32 owns one (b, chunk, 16-channel group). Within the chunk it
// processes 16 timesteps at a time:  D = L(16x16 lower-tri ones) @ Xtile + C,
// where C carries the running per-channel totals (cross-chunk prefix +
// previous tiles). Chained as 4x V_WMMA_F32_16X16X4_F32 per statistic.
// grid = (B_*NCHUNK*(C_/16))/8 = 512 blocks, block = 256 threads (8 waves).
// ---------------------------------------------------------------------------
__global__ void __launch_bounds__(256)
cln_apply(const float* __restrict__ x, const float* __restrict__ wgt,
          const float* __restrict__ bias, const float* __restrict__ pS,
          const float* __restrict__ pQ, float* __restrict__ out) {
    const int lane = threadIdx.x & 31;
    const int g    = blockIdx.x * 8 + (threadIdx.x >> 5);  // global wave id
    const int cg   = g & 31;                 // channel group (16 channels)
    const int k    = (g >> 5) & (NCHUNK - 1);
    const int b    = g >> 9;
    const int col  = lane & 15;
    const int hf   = lane >> 4;
    const int c    = cg * 16 + col;

    const float w   = wgt[c];
    const float bsv = bias[c];
    const float carryS = pS[(b * NCHUNK + k) * C_ + c];
    const float carryQ = pQ[(b * NCHUNK + k) * C_ + c];

    // A operand: lower-triangular ones, A[m][K] = (K <= m).
    // Layout (f32 16x4 slice): m = lane&15, K = 2*hf + v  (v = vgpr 0/1).
    v2f A[4];
    #pragma unroll
    for (int s4 = 0; s4 < 4; ++s4) {
        int kk = s4 * 4 + hf * 2;
        A[s4][0] = (kk + 0 <= col) ? 1.0f : 0.0f;
        A[s4][1] = (kk + 1 <= col) ? 1.0f : 0.0f;
    }

    v8f accS, accQ;
    #pragma unroll
    for (int i = 0; i < 8; ++i) { accS[i] = carryS; accQ[i] = carryQ; }

    const size_t baseBT = (size_t)b * T_;

    for (int tile = 0; tile < CHUNK_T / 16; ++tile) {
        const int t0 = k * CHUNK_T + tile * 16;

        // B operand: X[K][n], K = s4*4 + 2*hf + v, n = col.
        v2f Bv[4], B2[4];
        #pragma unroll
        for (int s4 = 0; s4 < 4; ++s4) {
            int row = s4 * 4 + hf * 2;
            float e0 = x[(baseBT + t0 + row    ) * C_ + c];
            float e1 = x[(baseBT + t0 + row + 1) * C_ + c];
            Bv[s4][0] = e0;       Bv[s4][1] = e1;
            B2[s4][0] = e0 * e0;  B2[s4][1] = e1 * e1;
        }

        // Cumulative sums: accS = L @ X + accS ; accQ = L @ (X*X) + accQ
        #pragma unroll
        for (int s4 = 0; s4 < 4; ++s4)
            accS = __builtin_amdgcn_wmma_f32_16x16x4_f32(
                false, A[s4], false, Bv[s4], (short)0, accS, false, false);
        #pragma unroll
        for (int s4 = 0; s4 < 4; ++s4)
            accQ = __builtin_amdgcn_wmma_f32_16x16x4_f32(
                false, A[s4], false, B2[s4], (short)0, accQ, false, false);

        // Normalize + store. C/D layout: row = r + 8*hf, col = lane&15.
        // rcp/rsq: single v_rcp_f32 / v_rsq_f32 (1-ulp) instead of the IEEE
        // div expansion — counts are small exact integers, so this is tight.
        #pragma unroll
        for (int r = 0; r < 8; ++r) {
            int t = t0 + r + hf * 8;
            float xv   = x[(baseBT + t) * C_ + c];   // L0/L2 hit (just loaded)
            float inv  = __builtin_amdgcn_rcpf((float)(t + 1));
            float mean = accS[r] * inv;
            float var  = fmaf(accQ[r], inv, -mean * mean);
            float o    = (xv - mean) * __builtin_amdgcn_rsqf(var + EPSV);
            out[(baseBT + t) * C_ + c] = fmaf(o, w, bsv);
        }

        // Carry = row 15 of D (VGPR 7, upper lane half), broadcast per column.
        float nS = __shfl(accS[7], 16 + col, 32);
        float nQ = __shfl(accQ[7], 16 + col, 32);
        #pragma unroll
        for (int i = 0; i < 8; ++i) { accS[i] = nS; accQ[i] = nQ; }
    }
}

extern "C" void kernel_launch(void* const* d_in, const int* in_sizes, int n_in,
                              void* d_out, int out_size, void* d_ws, size_t ws_size,
                              hipStream_t stream) {
    const float* x    = (const float*)d_in[0];
    const float* wgt  = (const float*)d_in[1];
    const float* bias = (const float*)d_in[2];
    float* out = (float*)d_out;

    // Workspace: chunk totals -> exclusive prefixes (rewritten every call).
    float* pS = (float*)d_ws;                  // B_*NCHUNK*C_ floats (256 KB)
    float* pQ = pS + (size_t)B_ * NCHUNK * C_; // another 256 KB

    cln_partials<<<B_ * NCHUNK, C_, 0, stream>>>(x, pS, pQ);
    cln_prefix<<<(B_ * C_) / 256, 256, 0, stream>>>(pS, pQ);
    cln_apply<<<(B_ * NCHUNK * (C_ / 16)) / 8, 256, 0, stream>>>(
        x, wgt, bias, pS, pQ, out);
}